// customGraphSAGE_13984413516240
// MI455X (gfx1250) — compile-verified
//
#include <hip/hip_runtime.h>

// Problem constants (fixed by the reference)
#define NNODES 50000
#define NEDGES 800000
#define ND 64          // node dim
#define ED 16          // edge dim
#define CIN 144        // 2*ND + ED
#define H1 256
#define H2 128
#define H3 64

// padded LDS row strides (odd -> conflict-free across 64 banks)
#define CINP (CIN + 1)   // 145
#define Z1P  (H1 + 1)    // 257
#define Z2P  (H2 + 1)    // 129
#define Z3P  (H3 + 1)    // 65
#define MT   32          // edges per head block (2 x 16-row WMMA tiles)
#define NWAVES 4         // waves cooperating on one head tile

typedef __attribute__((ext_vector_type(2))) float v2f;
typedef __attribute__((ext_vector_type(8))) float v8f;

// fp32 WMMA: D(16x16) = A(16x4) * B(4x16) + C
// A frag (ISA 7.12.2): lanes 0-15 hold M=lane, a[i] = K = k+i ; lanes 16-31 -> K = k+2+i
// B frag (KxN, row-striped): lanes 0-15 hold N=lane, b[i] = row K = k+i ; lanes 16-31 -> K = k+2+i
// C/D: element (M = r + 8*(lane>>4), N = lane&15) in vgpr r
__device__ __forceinline__ v8f wmma4(v2f a, v2f b, v8f c) {
  return __builtin_amdgcn_wmma_f32_16x16x4_f32(false, a, false, b, (short)0, c, false, false);
}

__global__ void k_zero(float* __restrict__ p, int n) {
  int i = blockIdx.x * blockDim.x + threadIdx.x;
  if (i < n) p[i] = 0.0f;
}

// One-time: segment-sum of edge features at dst + degree counts (L2-resident atomics)
__global__ void k_scatter_ef(const int* __restrict__ ei, const float* __restrict__ ef,
                             float* __restrict__ efagg, float* __restrict__ cnt) {
  int tid = blockIdx.x * blockDim.x + threadIdx.x;
  if (tid >= NEDGES * ED) return;
  int e = tid >> 4, f = tid & 15;
  int dst = ei[NEDGES + e];
  atomicAdd(&efagg[dst * ED + f], ef[tid]);
  if (f == 0) atomicAdd(&cnt[dst], 1.0f);
}

// Per layer: segment-sum of h[src] at dst
__global__ void k_scatter_x(const int* __restrict__ ei, const float* __restrict__ h,
                            float* __restrict__ xagg) {
  int tid = blockIdx.x * blockDim.x + threadIdx.x;
  if (tid >= NEDGES * ND) return;
  int e = tid >> 6, f = tid & 63;
  int src = ei[e];
  int dst = ei[NEDGES + e];
  atomicAdd(&xagg[dst * ND + f], h[src * ND + f]);
}

// One wave per block: 16-node tile, h_out = relu([h | xagg/cnt | efagg/cnt] @ W(144x64) + b)
// (LDS 9.3KB -> occupancy capped by the 32-workgroup/WGP limit, i.e. 8 waves/SIMD: fine.)
__global__ void __launch_bounds__(32)
k_sage_conv(const float* __restrict__ hcur, const float* __restrict__ xagg,
            const float* __restrict__ efagg, const float* __restrict__ cnt,
            const float* __restrict__ W, const float* __restrict__ bias,
            float* __restrict__ hout) {
  __shared__ float lin[16][CINP];
  const int lane = threadIdx.x;
  const int n0 = blockIdx.x * 16;

  for (int idx = lane; idx < 16 * CIN; idx += 32) {
    int r = idx / CIN, c = idx % CIN;
    int n = n0 + r;
    float inv = 1.0f / fmaxf(cnt[n], 1.0f);
    float v;
    if (c < ND)           v = hcur[n * ND + c];
    else if (c < 2 * ND)  v = xagg[n * ND + (c - ND)] * inv;
    else                  v = efagg[n * ED + (c - 2 * ND)] * inv;
    lin[r][c] = v;
  }
  __syncthreads();

  const int m = lane & 15;
  const int hi = lane >> 4;
  for (int nt = 0; nt < ND / 16; ++nt) {
    int col = nt * 16 + m;
    float bv = bias[col];
    v8f acc = {bv, bv, bv, bv, bv, bv, bv, bv};
    for (int k = 0; k < CIN; k += 4) {
      int ka = k + 2 * hi;
      v2f a, b;
      a.x = lin[m][ka];
      a.y = lin[m][ka + 1];
      b.x = W[ka * ND + col];
      b.y = W[(ka + 1) * ND + col];
      acc = wmma4(a, b, acc);
    }
#pragma unroll
    for (int r = 0; r < 8; ++r)
      hout[(n0 + r + 8 * hi) * ND + col] = fmaxf(acc[r], 0.0f);
  }
}

// 4 waves per block cooperate on one 32-edge tile of the fused 4-layer MLP head.
// Waves own disjoint output-column tiles (nt = w, w+NWAVES, ...): no duplicated weight
// traffic, shared A tile in LDS, and 4 waves amortize the 59.8KB LDS footprint
// (5 blocks x 4 waves = 20 waves/WGP = 5 waves/SIMD for latency hiding).
// Each B fragment still feeds two 16-row accumulators (2x reuse).
__global__ void __launch_bounds__(32 * NWAVES)
k_edge_head(const float* __restrict__ h, const int* __restrict__ ei,
            const float* __restrict__ ef,
            const float* __restrict__ W1, const float* __restrict__ b1,
            const float* __restrict__ W2, const float* __restrict__ b2,
            const float* __restrict__ W3, const float* __restrict__ b3,
            const float* __restrict__ W4, const float* __restrict__ b4,
            float* __restrict__ out) {
  __shared__ float smem[MT * Z1P + MT * CINP + MT * Z3P];  // 59776 B
  float* z1s = smem;                 // [MT][Z1P]
  float* pzs = smem + MT * Z1P;      // pair [MT][CINP]; aliased later as z2 [MT][Z2P]
  float* z3s = pzs + MT * CINP;      // [MT][Z3P]

  const int tid = threadIdx.x;
  const int lane = tid & 31;
  const int w = tid >> 5;            // wave id 0..3
  const int e0 = blockIdx.x * MT;

  // gather pair = [h[src] | h[dst] | ef] for 32 edges (all 128 threads)
  for (int idx = tid; idx < MT * CIN; idx += 32 * NWAVES) {
    int r = idx / CIN, c = idx % CIN;
    int e = e0 + r;
    float v;
    if (c < ND)           v = h[ei[e] * ND + c];
    else if (c < 2 * ND)  v = h[ei[NEDGES + e] * ND + (c - ND)];
    else                  v = ef[e * ED + (c - 2 * ND)];
    pzs[r * CINP + c] = v;
  }
  __syncthreads();

  const int m = lane & 15;
  const int hi = lane >> 4;

  // GEMM1: [32x144] @ [144x256] -> z1, relu   (4 col tiles per wave)
  for (int nt = w; nt < H1 / 16; nt += NWAVES) {
    int col = nt * 16 + m;
    float bv = b1[col];
    v8f acc0 = {bv, bv, bv, bv, bv, bv, bv, bv};
    v8f acc1 = acc0;
    for (int k = 0; k < CIN; k += 4) {
      int ka = k + 2 * hi;
      v2f b, a0, a1;
      b.x = W1[ka * H1 + col];
      b.y = W1[(ka + 1) * H1 + col];
      a0.x = pzs[m * CINP + ka];          a0.y = pzs[m * CINP + ka + 1];
      a1.x = pzs[(m + 16) * CINP + ka];   a1.y = pzs[(m + 16) * CINP + ka + 1];
      acc0 = wmma4(a0, b, acc0);
      acc1 = wmma4(a1, b, acc1);
    }
#pragma unroll
    for (int r = 0; r < 8; ++r) {
      z1s[(r + 8 * hi) * Z1P + col]      = fmaxf(acc0[r], 0.0f);
      z1s[(16 + r + 8 * hi) * Z1P + col] = fmaxf(acc1[r], 0.0f);
    }
  }
  __syncthreads();

  // GEMM2: [32x256] @ [256x128] -> z2 (aliased over pair), relu   (2 col tiles per wave)
  for (int nt = w; nt < H2 / 16; nt += NWAVES) {
    int col = nt * 16 + m;
    float bv = b2[col];
    v8f acc0 = {bv, bv, bv, bv, bv, bv, bv, bv};
    v8f acc1 = acc0;
    for (int k = 0; k < H1; k += 4) {
      int ka = k + 2 * hi;
      v2f b, a0, a1;
      b.x = W2[ka * H2 + col];
      b.y = W2[(ka + 1) * H2 + col];
      a0.x = z1s[m * Z1P + ka];          a0.y = z1s[m * Z1P + ka + 1];
      a1.x = z1s[(m + 16) * Z1P + ka];   a1.y = z1s[(m + 16) * Z1P + ka + 1];
      acc0 = wmma4(a0, b, acc0);
      acc1 = wmma4(a1, b, acc1);
    }
#pragma unroll
    for (int r = 0; r < 8; ++r) {
      pzs[(r + 8 * hi) * Z2P + col]      = fmaxf(acc0[r], 0.0f);
      pzs[(16 + r + 8 * hi) * Z2P + col] = fmaxf(acc1[r], 0.0f);
    }
  }
  __syncthreads();

  // GEMM3: [32x128] @ [128x64] -> z3, relu   (1 col tile per wave)
  for (int nt = w; nt < H3 / 16; nt += NWAVES) {
    int col = nt * 16 + m;
    float bv = b3[col];
    v8f acc0 = {bv, bv, bv, bv, bv, bv, bv, bv};
    v8f acc1 = acc0;
    for (int k = 0; k < H2; k += 4) {
      int ka = k + 2 * hi;
      v2f b, a0, a1;
      b.x = W3[ka * H3 + col];
      b.y = W3[(ka + 1) * H3 + col];
      a0.x = pzs[m * Z2P + ka];          a0.y = pzs[m * Z2P + ka + 1];
      a1.x = pzs[(m + 16) * Z2P + ka];   a1.y = pzs[(m + 16) * Z2P + ka + 1];
      acc0 = wmma4(a0, b, acc0);
      acc1 = wmma4(a1, b, acc1);
    }
#pragma unroll
    for (int r = 0; r < 8; ++r) {
      z3s[(r + 8 * hi) * Z3P + col]      = fmaxf(acc0[r], 0.0f);
      z3s[(16 + r + 8 * hi) * Z3P + col] = fmaxf(acc1[r], 0.0f);
    }
  }
  __syncthreads();

  // L4: 64 -> 1; one row per thread of wave 0
  if (tid < MT) {
    float s = b4[0];
    for (int k = 0; k < H3; ++k) s += z3s[tid * Z3P + k] * W4[k];
    out[e0 + tid] = s;
  }
}

extern "C" void kernel_launch(void* const* d_in, const int* in_sizes, int n_in,
                              void* d_out, int out_size, void* d_ws, size_t ws_size,
                              hipStream_t stream) {
  const float* x     = (const float*)d_in[0];
  const int*   ei    = (const int*)d_in[1];     // [2, E] int32
  const float* ef    = (const float*)d_in[2];
  const float* convW = (const float*)d_in[4];
  const float* convb = (const float*)d_in[5];
  const float* W1 = (const float*)d_in[6];  const float* b1 = (const float*)d_in[7];
  const float* W2 = (const float*)d_in[8];  const float* b2 = (const float*)d_in[9];
  const float* W3 = (const float*)d_in[10]; const float* b3 = (const float*)d_in[11];
  const float* W4 = (const float*)d_in[12]; const float* b4 = (const float*)d_in[13];
  float* out = (float*)d_out;

  // workspace layout (floats): cnt | efagg | xagg | h1 | h2  (~41.8 MB total)
  float* ws    = (float*)d_ws;
  float* cnt   = ws;                                   // NNODES
  float* efagg = cnt + NNODES;                         // NNODES*ED
  float* xagg  = efagg + (size_t)NNODES * ED;          // NNODES*ND
  float* h1    = xagg + (size_t)NNODES * ND;           // NNODES*ND
  float* h2    = h1 + (size_t)NNODES * ND;             // NNODES*ND

  auto cdiv = [](int a, int b) { return (a + b - 1) / b; };

  // cnt and efagg are contiguous: one zero pass
  k_zero<<<cdiv(NNODES * (1 + ED), 256), 256, 0, stream>>>(cnt, NNODES * (1 + ED));
  k_scatter_ef<<<cdiv(NEDGES * ED, 256), 256, 0, stream>>>(ei, ef, efagg, cnt);

  const float* hcur = x;
  float* houts[2] = {h1, h2};
  for (int layer = 0; layer < 2; ++layer) {
    k_zero<<<cdiv(NNODES * ND, 256), 256, 0, stream>>>(xagg, NNODES * ND);
    k_scatter_x<<<cdiv(NEDGES * ND, 256), 256, 0, stream>>>(ei, hcur, xagg);
    k_sage_conv<<<NNODES / 16, 32, 0, stream>>>(hcur, xagg, efagg, cnt, convW, convb,
                                                houts[layer]);
    hcur = houts[layer];
  }

  k_edge_head<<<NEDGES / MT, 32 * NWAVES, 0, stream>>>(h2, ei, ef, W1, b1, W2, b2, W3, b3,
                                                       W4, b4, out);
}